// Mamba_19241453486080
// MI455X (gfx1250) — compile-verified
//
#include <hip/hip_runtime.h>
#include <math.h>

typedef __attribute__((ext_vector_type(2))) float v2f;
typedef __attribute__((ext_vector_type(8))) float v8f;

#define BATCH   2
#define SEQLEN  4096
#define DMODEL  768
#define DINNER  1536
#define DSTATE  16
#define DCONV   4
#define DTRANK  48
#define XDBL_N  (DTRANK + 2 * DSTATE)   // 80
#define ML      (BATCH * SEQLEN)        // 8192

__device__ __forceinline__ float softplusf(float v) {
    return (v > 20.0f) ? v : log1pf(__expf(v));
}

// ---------------------------------------------------------------------------
// Big WMMA GEMM:  C[M,N] = A[M,K(lda)] * W[N,K]^T   (W row-major [N,K])
// wave tile 32x64 (2x4 frags), block = 8 waves = 64x256 tile.
// epilogue: 0 = plain store, 1 = softplus(acc + bias[col])
// Grid: (M/64, N/256), block 256. Requires K % 4 == 0.
// ---------------------------------------------------------------------------
__global__ __launch_bounds__(256) void gemm_big(
    const float* __restrict__ A, int lda,
    const float* __restrict__ W, int K,
    float* __restrict__ C, int ldc,
    const float* __restrict__ bias, int epilogue)
{
    const int tid  = threadIdx.x;
    const int lane = tid & 31;
    const int wav  = tid >> 5;
    const int kh   = lane >> 4;   // 0: lanes 0-15 (K=k..k+1), 1: lanes 16-31 (K=k+2..k+3)
    const int idx  = lane & 15;
    const int m0   = blockIdx.x * 64 + (wav & 1) * 32;
    const int n0   = blockIdx.y * 256 + (wav >> 1) * 64;

    v8f acc[2][4];
    #pragma unroll
    for (int i = 0; i < 2; i++)
        #pragma unroll
        for (int j = 0; j < 4; j++) acc[i][j] = {};

    const float* aRow0 = A + (size_t)(m0 + idx) * lda;
    const float* aRow1 = A + (size_t)(m0 + 16 + idx) * lda;
    const float* wRow0 = W + (size_t)(n0 +  0 + idx) * K;
    const float* wRow1 = W + (size_t)(n0 + 16 + idx) * K;
    const float* wRow2 = W + (size_t)(n0 + 32 + idx) * K;
    const float* wRow3 = W + (size_t)(n0 + 48 + idx) * K;

    for (int k = 0; k < K; k += 4) {
        const int kk = k + 2 * kh;
        v2f a0 = *(const v2f*)(aRow0 + kk);
        v2f a1 = *(const v2f*)(aRow1 + kk);
        v2f b0 = *(const v2f*)(wRow0 + kk);
        v2f b1 = *(const v2f*)(wRow1 + kk);
        v2f b2 = *(const v2f*)(wRow2 + kk);
        v2f b3 = *(const v2f*)(wRow3 + kk);
        acc[0][0] = __builtin_amdgcn_wmma_f32_16x16x4_f32(false, a0, false, b0, (short)0, acc[0][0], false, false);
        acc[0][1] = __builtin_amdgcn_wmma_f32_16x16x4_f32(false, a0, false, b1, (short)0, acc[0][1], false, false);
        acc[0][2] = __builtin_amdgcn_wmma_f32_16x16x4_f32(false, a0, false, b2, (short)0, acc[0][2], false, false);
        acc[0][3] = __builtin_amdgcn_wmma_f32_16x16x4_f32(false, a0, false, b3, (short)0, acc[0][3], false, false);
        acc[1][0] = __builtin_amdgcn_wmma_f32_16x16x4_f32(false, a1, false, b0, (short)0, acc[1][0], false, false);
        acc[1][1] = __builtin_amdgcn_wmma_f32_16x16x4_f32(false, a1, false, b1, (short)0, acc[1][1], false, false);
        acc[1][2] = __builtin_amdgcn_wmma_f32_16x16x4_f32(false, a1, false, b2, (short)0, acc[1][2], false, false);
        acc[1][3] = __builtin_amdgcn_wmma_f32_16x16x4_f32(false, a1, false, b3, (short)0, acc[1][3], false, false);
    }

    #pragma unroll
    for (int i = 0; i < 2; i++) {
        #pragma unroll
        for (int j = 0; j < 4; j++) {
            const int col = n0 + j * 16 + idx;
            const float bv = epilogue ? bias[col] : 0.0f;
            #pragma unroll
            for (int r = 0; r < 8; r++) {
                const int row = m0 + i * 16 + kh * 8 + r;   // D layout: VGPR r -> M=r / r+8
                float v = acc[i][j][r];
                if (epilogue) v = softplusf(v + bv);
                C[(size_t)row * ldc + col] = v;
            }
        }
    }
}

// ---------------------------------------------------------------------------
// x_dbl GEMM: C[ML,80] = A[ML,1536] * W_x[80,1536]^T
// one wave = 16 M rows x 5 N-tiles (entire N=80), block = 8 waves = 128 rows.
// Grid: ML/128, block 256.
// ---------------------------------------------------------------------------
__global__ __launch_bounds__(256) void gemm_xdbl(
    const float* __restrict__ A,
    const float* __restrict__ W,
    float* __restrict__ C)
{
    const int tid  = threadIdx.x;
    const int lane = tid & 31;
    const int wav  = tid >> 5;
    const int kh   = lane >> 4;
    const int idx  = lane & 15;
    const int m0   = (blockIdx.x * 8 + wav) * 16;

    v8f acc[5];
    #pragma unroll
    for (int j = 0; j < 5; j++) acc[j] = {};

    const float* aRow = A + (size_t)(m0 + idx) * DINNER;
    const float* wRow[5];
    #pragma unroll
    for (int j = 0; j < 5; j++) wRow[j] = W + (size_t)(j * 16 + idx) * DINNER;

    for (int k = 0; k < DINNER; k += 4) {
        const int kk = k + 2 * kh;
        v2f a = *(const v2f*)(aRow + kk);
        #pragma unroll
        for (int j = 0; j < 5; j++) {
            v2f b = *(const v2f*)(wRow[j] + kk);
            acc[j] = __builtin_amdgcn_wmma_f32_16x16x4_f32(false, a, false, b, (short)0, acc[j], false, false);
        }
    }

    #pragma unroll
    for (int j = 0; j < 5; j++) {
        const int col = j * 16 + idx;
        #pragma unroll
        for (int r = 0; r < 8; r++) {
            const int row = m0 + kh * 8 + r;
            C[(size_t)row * XDBL_N + col] = acc[j][r];
        }
    }
}

// ---------------------------------------------------------------------------
// Depthwise causal conv1d (width 4) + SiLU.  x lives in xz[..., 0:DINNER].
// ---------------------------------------------------------------------------
__global__ __launch_bounds__(256) void conv_silu(
    const float* __restrict__ xz,
    const float* __restrict__ cw,
    const float* __restrict__ cb,
    float* __restrict__ xc)
{
    const size_t gid = (size_t)blockIdx.x * 256 + threadIdx.x;
    if (gid >= (size_t)ML * DINNER) return;
    const int d  = (int)(gid % DINNER);
    const size_t bl = gid / DINNER;          // b*L + l
    const int l  = (int)(bl % SEQLEN);

    float acc = cb[d];
    #pragma unroll
    for (int j = 0; j < DCONV; j++) {
        const int ls = l - (DCONV - 1) + j;
        if (ls >= 0)
            acc += cw[d * DCONV + j] * xz[(bl + (size_t)(ls - l)) * (size_t)(2 * DINNER) + d];
    }
    xc[bl * (size_t)DINNER + d] = acc / (1.0f + __expf(-acc));  // SiLU
}

// ---------------------------------------------------------------------------
// Selective scan. 16 lanes = 1 channel (one state per lane), wave32 = 2 channels.
// Block 256 = 16 channels.  Grid: (BATCH*DINNER)/16 = 192.
// y is written in place over xc (read-before-write within the wave).
// ---------------------------------------------------------------------------
__global__ __launch_bounds__(256) void scan_kernel(
    const float* __restrict__ xz,
    float* __restrict__ xc,              // in: conv x, out: gated y
    const float* __restrict__ xdbl,
    const float* __restrict__ dtb,
    const float* __restrict__ A_log,
    const float* __restrict__ Dv)
{
    const int tid = threadIdx.x;
    const int s   = tid & 15;                      // state index
    const int ch  = blockIdx.x * 16 + (tid >> 4);  // global channel = b*DINNER + d
    const int d   = ch % DINNER;
    const int b   = ch / DINNER;

    const float Aval = -__expf(A_log[d * DSTATE + s]);
    const float Dd   = Dv[d];

    const float* dt_p = dtb  + (size_t)b * SEQLEN * DINNER + d;
    float*       x_p  = xc   + (size_t)b * SEQLEN * DINNER + d;
    const float* z_p  = xz   + (size_t)b * SEQLEN * (2 * DINNER) + DINNER + d;
    const float* bc_p = xdbl + (size_t)b * SEQLEN * XDBL_N + DTRANK;

    float h = 0.0f;
    for (int l = 0; l < SEQLEN; l++) {
        const float dt_t = dt_p[(size_t)l * DINNER];
        const float x_t  = x_p [(size_t)l * DINNER];
        const float Bv   = bc_p[(size_t)l * XDBL_N + s];
        const float Cv   = bc_p[(size_t)l * XDBL_N + DSTATE + s];

        h = __expf(dt_t * Aval) * h + (dt_t * x_t) * Bv;

        float p = h * Cv;
        p += __shfl_xor(p, 8, 16);
        p += __shfl_xor(p, 4, 16);
        p += __shfl_xor(p, 2, 16);
        p += __shfl_xor(p, 1, 16);

        if (s == 0) {
            const float z_t = z_p[(size_t)l * (2 * DINNER)];
            float y = p + x_t * Dd;
            y *= z_t / (1.0f + __expf(-z_t));   // * silu(z)
            x_p[(size_t)l * DINNER] = y;
        }
    }
}

// ---------------------------------------------------------------------------
extern "C" void kernel_launch(void* const* d_in, const int* in_sizes, int n_in,
                              void* d_out, int out_size, void* d_ws, size_t ws_size,
                              hipStream_t stream)
{
    const float* hs     = (const float*)d_in[0];
    const float* W_in   = (const float*)d_in[1];
    const float* conv_w = (const float*)d_in[2];
    const float* conv_b = (const float*)d_in[3];
    const float* W_x    = (const float*)d_in[4];
    const float* W_dt   = (const float*)d_in[5];
    const float* b_dt   = (const float*)d_in[6];
    const float* A_log  = (const float*)d_in[7];
    const float* Dv     = (const float*)d_in[8];
    const float* W_out  = (const float*)d_in[9];
    float* out = (float*)d_out;

    float* ws   = (float*)d_ws;
    float* xz   = ws;                               // ML * 3072
    float* xc   = xz   + (size_t)ML * 2 * DINNER;   // ML * 1536 (conv x, then y)
    float* xdbl = xc   + (size_t)ML * DINNER;       // ML * 80
    float* dtb  = xdbl + (size_t)ML * XDBL_N;       // ML * 1536

    dim3 blk(256);

    // 1) xz = hs @ W_in^T                (8192 x 3072, K=768)
    gemm_big<<<dim3(ML / 64, (2 * DINNER) / 256), blk, 0, stream>>>(
        hs, DMODEL, W_in, DMODEL, xz, 2 * DINNER, nullptr, 0);

    // 2) depthwise conv + SiLU
    {
        size_t total = (size_t)ML * DINNER;
        conv_silu<<<dim3((unsigned)((total + 255) / 256)), blk, 0, stream>>>(
            xz, conv_w, conv_b, xc);
    }

    // 3) x_dbl = x @ W_x^T               (8192 x 80, K=1536)
    gemm_xdbl<<<dim3(ML / 128), blk, 0, stream>>>(xc, W_x, xdbl);

    // 4) dt = softplus(dt_lo @ W_dt^T + b_dt)   (8192 x 1536, K=48, lda=80)
    gemm_big<<<dim3(ML / 64, DINNER / 256), blk, 0, stream>>>(
        xdbl, XDBL_N, W_dt, DTRANK, dtb, DINNER, b_dt, 1);

    // 5) selective scan (+ D skip + SiLU(z) gate), y overwrites xc
    scan_kernel<<<dim3((BATCH * DINNER) / 16), blk, 0, stream>>>(
        xz, xc, xdbl, dtb, A_log, Dv);

    // 6) out = y @ W_out^T               (8192 x 768, K=1536)
    gemm_big<<<dim3(ML / 64, DMODEL / 256), blk, 0, stream>>>(
        xc, DINNER, W_out, DINNER, out, DMODEL, nullptr, 0);
}